// CustomLoss_23072564314320
// MI455X (gfx1250) — compile-verified
//
#include <hip/hip_runtime.h>

typedef unsigned __int128 u128;
typedef __attribute__((ext_vector_type(2))) float v2f;
typedef __attribute__((ext_vector_type(8))) float v8f;

#define G 10
#define NCELL 100
#define BIG_F 1000000000.0f
#define WEIGHT 20000.0f
#define GAP_WEIGHT 3000.0f
#define WEIGHT_WEIGHT 1.1f

// ---- compile-time 128-bit grid masks -------------------------------------
constexpr u128 colmask(int c) {
    u128 m = 0;
    for (int r = 0; r < G; ++r) m |= ((u128)1) << (r * G + c);
    return m;
}
constexpr u128 FULL100 = (((u128)1) << NCELL) - 1;
constexpr u128 NOTC0   = FULL100 & ~colmask(0);     // cells with col > 0
constexpr u128 NOTC9   = FULL100 & ~colmask(G - 1); // cells with col < 9

// 8-connected dilation, clipped to the 10x10 grid
__device__ __forceinline__ u128 dil8(u128 m) {
    u128 w = m & NOTC0;   // can move left
    u128 e = m & NOTC9;   // can move right
    u128 r = m
           | (w >> 1) | (e << 1)               // W, E
           | (m >> G) | (m << G)               // N, S
           | (w >> (G + 1)) | (w << (G - 1))   // NW, SW
           | (e >> (G - 1)) | (e << (G + 1));  // NE, SE
    return r & FULL100;
}

// 4-connected (Manhattan) dilation, unrestricted by fg
__device__ __forceinline__ u128 dil4(u128 m) {
    u128 w = m & NOTC0;
    u128 e = m & NOTC9;
    return (m | (w >> 1) | (e << 1) | (m >> G) | (m << G)) & FULL100;
}

__device__ __forceinline__ int popc128(u128 m) {
    return __popcll((unsigned long long)m) +
           __popcll((unsigned long long)(m >> 64));
}

// flood fill from seed bit within fg; returns empty if seed not in fg
__device__ __forceinline__ u128 flood(u128 fg, int idx) {
    if (!((fg >> idx) & (u128)1)) return (u128)0;
    u128 comp = ((u128)1) << idx;
    #pragma unroll 1
    for (int it = 0; it < NCELL; ++it) {
        u128 nxt = dil8(comp) & fg;
        if (nxt == comp) break;
        comp = nxt;
    }
    return comp;
}

// ---- kernel 1: one wave32 per batch item ---------------------------------
__global__ void loss_per_item(const float* __restrict__ result,
                              const float* __restrict__ weight,
                              const int*   __restrict__ points,
                              float* __restrict__ losses, int B) {
    const int lane = threadIdx.x & 31;
    const int wave = threadIdx.x >> 5;
    const int item = blockIdx.x * 8 + wave;
    if (item >= B) return;

    const size_t base = (size_t)item * NCELL;

    // coalesced loads: lane owns cells lane, lane+32, lane+64, lane+96
    float rsum = 0.0f, rwsum = 0.0f;
    unsigned m[4];
    #pragma unroll
    for (int j = 0; j < 4; ++j) {
        int c = lane + 32 * j;
        bool in = (c < NCELL);
        float x = in ? result[base + c] : 0.0f;
        float y = in ? weight[base + c] : 0.0f;
        rsum  += x;
        rwsum += x * y;
        m[j] = __builtin_amdgcn_ballot_w32(in && (rintf(x) > 0.5f));
    }
    u128 fg = ((u128)m[0]) | ((u128)m[1] << 32) | ((u128)m[2] << 64) | ((u128)m[3] << 96);

    // wave-wide sums (deterministic tree)
    #pragma unroll
    for (int off = 16; off > 0; off >>= 1) {
        rsum  += __shfl_xor(rsum,  off, 32);
        rwsum += __shfl_xor(rwsum, off, 32);
    }

    // seed points
    const int p0r = points[item * 4 + 0], p0c = points[item * 4 + 1];
    const int p1r = points[item * 4 + 2], p1c = points[item * 4 + 3];
    const int idx0 = p0r * G + p0c;
    const int idx1 = p1r * G + p1c;

    const float r0 = result[base + idx0];
    const float r1 = result[base + idx1];

    // components containing the two seed points (empty if seed not fg)
    const u128 comp0 = flood(fg, idx0);
    const u128 comp1 = flood(fg, idx1);
    const bool both_fg = (comp0 != (u128)0) && (comp1 != (u128)0);

    // min Manhattan distance between comp0 and comp1 via BFS dilation (<=18)
    float min_dist = 0.0f;
    if (both_fg && !(comp0 & comp1)) {
        u128 t = comp1;
        min_dist = BIG_F;
        #pragma unroll 1
        for (int d = 1; d <= 2 * (G - 1); ++d) {
            t = dil4(t);
            if (t & comp0) { min_dist = (float)d; break; }
        }
    }

    const int n_start = both_fg ? popc128(comp0) : 0;
    const float manhattan = (float)(abs(p1r - p0r) + abs(p1c - p0c));
    const float soa_cells_inv = 100.0f - rsum;

    const float pen = (2.0f - r0 - r1) * WEIGHT;
    const float loss_start = ((rintf(r0) == 0.0f) || (r1 == 0.0f)) ? pen : 0.0f;
    const float gap_loss = both_fg ? (min_dist * soa_cells_inv * GAP_WEIGHT) : pen;
    const float csp = rwsum * WEIGHT_WEIGHT * fabsf(manhattan - (float)n_start);

    if (lane == 0) losses[item] = loss_start + gap_loss + csp;
}

// ---- kernel 2: exact fp32 WMMA reduction of the per-item losses ----------
// V_WMMA_F32_16X16X4_F32 with B = ones folds 64 fp32 values per instruction
// into 16 fp32 row accumulators (every column identical). fp32 in/out so
// losses (~1e7) don't overflow the way f16 would. Main loop runs with no
// divergence at all -> EXEC is all-ones around every WMMA (ISA requirement),
// unconditional b64 loads, two independent accumulators to break the RAW
// chain between consecutive WMMAs. Tail (B % 128) is scalar, non-WMMA.
__device__ __forceinline__ v8f wmma_accum(v2f a, v2f b, v8f c) {
    return __builtin_amdgcn_wmma_f32_16x16x4_f32(
        false, a, false, b, (short)0, c, false, false);
}

__global__ void reduce_mean(const float* __restrict__ vals,
                            float* __restrict__ out, int B) {
    __shared__ float partial[8];
    const int lane = threadIdx.x & 31;
    const int wave = threadIdx.x >> 5;

    v8f c0 = {}, c1 = {};
    v2f ones; ones[0] = 1.0f; ones[1] = 1.0f;

    const int nchunks = B >> 7;            // whole 128-element chunks
    const v2f* v2 = (const v2f*)vals;      // d_ws is 8B-aligned

    #pragma unroll 1
    for (int ch = wave; ch < nchunks; ch += 8) {
        const int p = ch * 64;             // v2f index of chunk base
        v2f a0 = v2[p + lane];             // elements [128*ch      , +64)
        v2f a1 = v2[p + 32 + lane];        // elements [128*ch + 64 , +64)
        c0 = wmma_accum(a0, ones, c0);
        c1 = wmma_accum(a1, ones, c1);
    }

    // scalar tail: elements [nchunks*128, B)  (empty when B % 128 == 0)
    float tail = 0.0f;
    if (wave == 0) {
        #pragma unroll 1
        for (int i = (nchunks << 7) + lane; i < B; i += 32) tail += vals[i];
    }
    #pragma unroll
    for (int off = 16; off > 0; off >>= 1) tail += __shfl_xor(tail, off, 32);

    // combine accumulators; each lane's column holds identical row-sums, so
    // after summing the 8 VGPRs and folding lane L with lane L+16 every lane
    // holds this wave's full partial sum.
    v8f c = c0 + c1;
    float s = c[0] + c[1] + c[2] + c[3] + c[4] + c[5] + c[6] + c[7];
    s += __shfl_xor(s, 16, 32);

    if (lane == 0) partial[wave] = s + tail;
    __syncthreads();
    if (threadIdx.x == 0) {
        float t = 0.0f;
        for (int k = 0; k < 8; ++k) t += partial[k];
        out[0] = t / (float)B;
    }
}

extern "C" void kernel_launch(void* const* d_in, const int* in_sizes, int n_in,
                              void* d_out, int out_size, void* d_ws, size_t ws_size,
                              hipStream_t stream) {
    const float* result = (const float*)d_in[0]; // result_given   [B,1,10,10] f32
    const int*   points = (const int*)  d_in[1]; // points_given   [B,2,2]     i32
    const float* weight = (const float*)d_in[2]; // weightmatrix   [B,1,10,10] f32
    float* out    = (float*)d_out;
    float* losses = (float*)d_ws;                // B floats of scratch

    const int B = in_sizes[0] / NCELL;
    const int blocks = (B + 7) / 8;              // 8 waves (items) per 256-thread block

    loss_per_item<<<blocks, 256, 0, stream>>>(result, weight, points, losses, B);
    reduce_mean<<<1, 256, 0, stream>>>(losses, out, B);
}